// PatternLoss_3D_44152263803104
// MI455X (gfx1250) — compile-verified
//
#include <hip/hip_runtime.h>
#include <hip/hip_bf16.h>
#include <stdint.h>

typedef __attribute__((ext_vector_type(2))) float v2f;
typedef __attribute__((ext_vector_type(8))) float v8f;

static constexpr float THRESH = (float)((127.0 / 255.0 - 0.5) / 0.5);

// ---------------- zero the 6x512 global histograms ----------------
__global__ void k_zero_hist(uint32_t* __restrict__ hist) {
    int i = blockIdx.x * blockDim.x + threadIdx.x;
    if (i < 6 * 512) hist[i] = 0u;
}

// ---------------- binarize + pack along w (bit pos = w%32) ----------------
// one thread per element of both tensors; wave32 ballot builds the mask
__global__ void k_pack_w(const float* __restrict__ inp, const float* __restrict__ tgt,
                         uint32_t* __restrict__ packW) {
    uint32_t idx = blockIdx.x * 256u + threadIdx.x;      // 0 .. 2*2^25-1
    uint32_t t   = idx >> 25;                            // tensor select
    uint32_t e   = idx & 33554431u;                      // element in tensor
    const float* src = t ? tgt : inp;
    float x = src[e];
    uint32_t m = __builtin_amdgcn_ballot_w32(x >= THRESH);
    if ((threadIdx.x & 31u) == 0u)
        packW[t * 1048576u + (e >> 5)] = m;              // bit (e%32) = lane
}

// ---------------- repack along h (bit pos = h%32), from packW ----------------
__global__ void k_pack_h(const uint32_t* __restrict__ packW, uint32_t* __restrict__ packH) {
    uint32_t idx  = blockIdx.x * 256u + threadIdx.x;     // 2^26 threads
    uint32_t lane = idx & 31u;
    uint32_t k    = (idx >> 5) & 7u;
    uint32_t w    = (idx >> 8) & 255u;
    uint32_t c    = (idx >> 16) & 255u;
    uint32_t n    = (idx >> 24) & 1u;
    uint32_t t    = (idx >> 25) & 1u;
    uint32_t h    = k * 32u + lane;
    uint32_t word = packW[t * 1048576u + ((n * 256u + c) * 256u + h) * 8u + (w >> 5)];
    uint32_t bit  = (word >> (w & 31u)) & 1u;
    uint32_t m = __builtin_amdgcn_ballot_w32(bit != 0u);
    if (lane == 0u)
        packH[t * 1048576u + ((n * 256u + c) * 256u + w) * 8u + k] = m;
}

// ---------------- 3x3 pattern-code histogram over packed rows ----------------
// rows are 256 bits (8 u32). Window spans 3 rows (stride istride u32) and
// bit positions q..q+2 within the row. Each wave handles one (n,mid,low)
// position; lane L extracts codes for q = 8L .. 8L+7 (lane 31: 6 codes).
__global__ void k_hist(const uint32_t* __restrict__ pk, uint32_t* __restrict__ ghist,
                       uint32_t lowMul, uint32_t midMul, uint32_t istride) {
    __shared__ uint32_t lh[512];
    for (int i = threadIdx.x; i < 512; i += blockDim.x) lh[i] = 0u;
    __syncthreads();

    const uint32_t P = 2u * 256u * 254u;                 // positions
    uint32_t lane   = threadIdx.x & 31u;
    uint32_t wave   = blockIdx.x * (blockDim.x >> 5) + (threadIdx.x >> 5);
    uint32_t nwaves = gridDim.x * (blockDim.x >> 5);
    uint32_t kw = lane >> 2;                             // word index 0..7
    uint32_t sh = (lane & 3u) * 8u;                      // bit offset in word
    int ncodes = (lane == 31u) ? 6 : 8;

    for (uint32_t p = wave; p < P; p += nwaves) {
        uint32_t n   = p / (256u * 254u);
        uint32_t r   = p - n * (256u * 254u);
        uint32_t mid = r / 254u;
        uint32_t low = r - mid * 254u;
        uint32_t base = n * 524288u + mid * midMul + low * lowMul;
        uint64_t v[3];
#pragma unroll
        for (int i = 0; i < 3; ++i) {
            uint32_t rb = base + (uint32_t)i * istride + kw;
            uint32_t lo = pk[rb];
            uint32_t hi = (kw < 7u) ? pk[rb + 1u] : 0u;
            v[i] = ((((uint64_t)hi) << 32) | (uint64_t)lo) >> sh;
        }
        for (int tt = 0; tt < ncodes; ++tt) {
            uint32_t code = (uint32_t)((v[0] >> tt) & 7u)
                          | ((uint32_t)((v[1] >> tt) & 7u) << 3)
                          | ((uint32_t)((v[2] >> tt) & 7u) << 6);
            atomicAdd(&lh[code], 1u);
        }
    }
    __syncthreads();
    for (int i = threadIdx.x; i < 512; i += blockDim.x)
        if (lh[i]) atomicAdd(&ghist[i], lh[i]);
}

// ---------------- final: sum of squared histogram diffs via WMMA ----------------
// single wave32; 3*512 squared diffs summed with v_wmma_f32_16x16x4_f32
// (A = values, B = ones, C accumulates row sums; every column of D holds the total)
__global__ void k_final(const uint32_t* __restrict__ hist, float* __restrict__ out) {
    __shared__ float sq[1536];
    uint32_t L = threadIdx.x;                            // 0..31
    for (uint32_t j = L; j < 1536u; j += 32u) {
        uint32_t d = j >> 9, b = j & 511u;
        float diff = (float)((int)hist[(2u * d) * 512u + b] -
                             (int)hist[(2u * d + 1u) * 512u + b]);
        sq[j] = diff * diff;
    }
    __syncthreads();

    v8f acc = {};
    v2f ones; ones.x = 1.0f; ones.y = 1.0f;
    uint32_t m  = (L < 16u) ? L : (L - 16u);             // A 16x4 f32 layout
    uint32_t ko = (L < 16u) ? 0u : 2u;                   // lanes 16-31 hold K=2,3
#pragma unroll
    for (int c = 0; c < 24; ++c) {                       // 24 * 64 = 1536 values
        uint32_t base = (uint32_t)c * 64u;
        v2f a;
        a.x = sq[base + m * 4u + ko];
        a.y = sq[base + m * 4u + ko + 1u];
        acc = __builtin_amdgcn_wmma_f32_16x16x4_f32(false, a, false, ones,
                                                    (short)0, acc, false, false);
    }
    float partial = acc[0] + acc[1] + acc[2] + acc[3] + acc[4] + acc[5] + acc[6] + acc[7];
    float other   = __shfl_xor(partial, 16, 32);         // rows 8-15 live in lane^16
    float total   = partial + other;
    if (L == 0u) {
        constexpr double T = 33032192.0;                 // codes per (dir,tensor)
        constexpr double scale = 1.0 / (3.0 * 512.0 * 512.0 * T * T);
        out[0] = (float)((double)total * scale);
    }
}

extern "C" void kernel_launch(void* const* d_in, const int* in_sizes, int n_in,
                              void* d_out, int out_size, void* d_ws, size_t ws_size,
                              hipStream_t stream) {
    (void)in_sizes; (void)n_in; (void)out_size; (void)ws_size;
    const float* inp = (const float*)d_in[0];
    const float* tgt = (const float*)d_in[1];
    uint32_t* ws    = (uint32_t*)d_ws;
    uint32_t* hist  = ws;                    // 6*512 u32
    uint32_t* packW = ws + 3072u;            // 2 * 1048576 u32 (w-packed, in/tgt)
    uint32_t* packH = packW + 2097152u;      // 2 * 1048576 u32 (h-packed, in/tgt)
    float* out = (float*)d_out;

    k_zero_hist<<<12, 256, 0, stream>>>(hist);
    k_pack_w<<<262144, 256, 0, stream>>>(inp, tgt, packW);
    k_pack_h<<<262144, 256, 0, stream>>>(packW, packH);

    // dir xy: windows over (h,w); rows stride 8 (h), packW
    k_hist<<<1024, 256, 0, stream>>>(packW,             hist + 0 * 512, 8u,    2048u, 8u);
    k_hist<<<1024, 256, 0, stream>>>(packW + 1048576u,  hist + 1 * 512, 8u,    2048u, 8u);
    // dir yz: windows over (c,w); rows stride 2048 (c), packW
    k_hist<<<1024, 256, 0, stream>>>(packW,             hist + 2 * 512, 2048u, 8u,    2048u);
    k_hist<<<1024, 256, 0, stream>>>(packW + 1048576u,  hist + 3 * 512, 2048u, 8u,    2048u);
    // dir zx: windows over (c,h); rows stride 2048 (c), packH
    k_hist<<<1024, 256, 0, stream>>>(packH,             hist + 4 * 512, 2048u, 8u,    2048u);
    k_hist<<<1024, 256, 0, stream>>>(packH + 1048576u,  hist + 5 * 512, 2048u, 8u,    2048u);

    k_final<<<1, 32, 0, stream>>>(hist, out);
}